// BVH_69106023793126
// MI455X (gfx1250) — compile-verified
//
#include <hip/hip_runtime.h>

#define NTRI   1024
#define NPTS   16384
#define NSLICE 4                    // triangle-dim split for machine fill
#define STRI   (NTRI / NSLICE)      // 256 triangles per slice
#define BLK    128                  // 4 waves of 32 on CDNA5

typedef int v4i __attribute__((vector_size(16)));
typedef __attribute__((address_space(1))) v4i* g_v4i_p;   // global
typedef __attribute__((address_space(3))) v4i* l_v4i_p;   // LDS

// Workspace layout (float4 units):
//   [0,            3*NTRI)              triangle records (48 KB)
//   [3*NTRI,       3*NTRI+NSLICE*NPTS)  per-(slice,point) candidates (1 MB)
#define WS_CAND_BASE (3 * NTRI)

// Triangle record (3 x float4 = 48B):
//  q0 = (a.x, a.y, a.z, ab.ab)
//  q1 = (ab.x, ab.y, ab.z, ab.ac)
//  q2 = (ac.x, ac.y, ac.z, ac.ac)

__global__ __launch_bounds__(256) void tri_prep_kernel(const float* __restrict__ tri,
                                                       float4* __restrict__ ws) {
    int t = blockIdx.x * blockDim.x + threadIdx.x;
    if (t >= NTRI) return;
    const float* s = tri + t * 9;
    float ax = s[0], ay = s[1], az = s[2];
    float bx = s[3], by = s[4], bz = s[5];
    float cx = s[6], cy = s[7], cz = s[8];
    float abx = bx - ax, aby = by - ay, abz = bz - az;
    float acx = cx - ax, acy = cy - ay, acz = cz - az;
    float ab_ab = abx * abx + aby * aby + abz * abz;
    float ab_ac = abx * acx + aby * acy + abz * acz;
    float ac_ac = acx * acx + acy * acy + acz * acz;
    float4* o = ws + t * 3;
    o[0] = make_float4(ax, ay, az, ab_ab);
    o[1] = make_float4(abx, aby, abz, ab_ac);
    o[2] = make_float4(acx, acy, acz, ac_ac);
}

__device__ __forceinline__ float rcp_safe(float y) {
    float d = (y == 0.0f) ? 1.0f : y;   // jnp: x / where(y==0, 1, y)
    return __builtin_amdgcn_rcpf(d);
}

// One block = 128 points x one 256-triangle slice. Emits best candidate per point.
__global__ __launch_bounds__(BLK) void slice_kernel(const float* __restrict__ pts,
                                                    const float4* __restrict__ ws,
                                                    float4* __restrict__ cand) {
    __shared__ float4 ltri[STRI * 3];   // 12 KB -> many co-resident WGs per 320KB WGP

    const int tid   = threadIdx.x;
    const int slice = blockIdx.y;
    const int fbase = slice * STRI;
    const float4* gtri = ws + fbase * 3;

    // ---- CDNA5 async global->LDS staging of this slice ----
    for (int i = tid; i < STRI * 3; i += BLK) {
#if __has_builtin(__builtin_amdgcn_global_load_async_to_lds_b128)
        __builtin_amdgcn_global_load_async_to_lds_b128(
            (g_v4i_p)(gtri + i), (l_v4i_p)(&ltri[i]), 0, 0);
#else
        ltri[i] = gtri[i];
#endif
    }
#if __has_builtin(__builtin_amdgcn_s_wait_asynccnt)
    __builtin_amdgcn_s_wait_asynccnt(0);
#else
    asm volatile("s_wait_asynccnt 0" ::: "memory");
#endif
    __syncthreads();

    const int pi = blockIdx.x * BLK + tid;
    const float px = pts[3 * pi + 0];
    const float py = pts[3 * pi + 1];
    const float pz = pts[3 * pi + 2];

    float best = 3.402823466e+38f;
    float bv = 0.0f, bw = 0.0f;
    int   bf = fbase;

#pragma unroll 2
    for (int f = 0; f < STRI; ++f) {
        // wave-uniform LDS address -> conflict-free broadcast
        float4 q0 = ltri[f * 3 + 0];
        float4 q1 = ltri[f * 3 + 1];
        float4 q2 = ltri[f * 3 + 2];

        float apx = px - q0.x, apy = py - q0.y, apz = pz - q0.z;
        float d1 = q1.x * apx + q1.y * apy + q1.z * apz;   // ab . ap
        float d2 = q2.x * apx + q2.y * apy + q2.z * apz;   // ac . ap
        float d3 = d1 - q0.w;                              // d1 - ab.ab
        float d4 = d2 - q1.w;                              // d2 - ab.ac
        float d5 = d1 - q1.w;                              // d1 - ab.ac
        float d6 = d2 - q2.w;                              // d2 - ac.ac

        float vc = d1 * d4 - d3 * d2;
        float vb = d5 * d2 - d1 * d6;
        float va = d3 * d6 - d5 * d4;
        float rden = rcp_safe(va + vb + vc);
        float v = vb * rden;
        float w = vc * rden;

        // exact jnp.where cascade order (later masks override earlier)
        float e43 = d4 - d3, e56 = d5 - d6;
        float t = e43 * rcp_safe(e43 + e56);
        if (va <= 0.0f && e43 >= 0.0f && e56 >= 0.0f) { v = 1.0f - t; w = t; }
        if (vb <= 0.0f && d2 >= 0.0f && d6 <= 0.0f)   { v = 0.0f; w = d2 * rcp_safe(d2 - d6); }
        if (vc <= 0.0f && d1 >= 0.0f && d3 <= 0.0f)   { v = d1 * rcp_safe(d1 - d3); w = 0.0f; }
        if (d6 >= 0.0f && d5 <= d6)                   { v = 0.0f; w = 1.0f; }
        if (d3 >= 0.0f && d4 <= d3)                   { v = 1.0f; w = 0.0f; }
        if (d1 <= 0.0f && d2 <= 0.0f)                 { v = 0.0f; w = 0.0f; }

        // dist2 = | ap - v*ab - w*ac |^2
        float ex = apx - v * q1.x - w * q2.x;
        float ey = apy - v * q1.y - w * q2.y;
        float ez = apz - v * q1.z - w * q2.z;
        float dd = ex * ex + ey * ey + ez * ez;

        if (dd < best) { best = dd; bv = v; bw = w; bf = fbase + f; }  // first-index ties
    }

    cand[slice * NPTS + pi] = make_float4(best, bv, bw, __int_as_float(bf));
}

// Fold NSLICE candidates per point, reconstruct closest point, emit outputs.
__global__ __launch_bounds__(256) void reduce_kernel(const float4* __restrict__ ws,
                                                     float* __restrict__ out) {
    const int pi = blockIdx.x * blockDim.x + threadIdx.x;
    if (pi >= NPTS) return;
    const float4* cand = ws + WS_CAND_BASE;

    float4 best = cand[pi];
#pragma unroll
    for (int s = 1; s < NSLICE; ++s) {
        float4 c = cand[s * NPTS + pi];
        if (c.x < best.x) best = c;      // slice order + strict < => global first-min
    }
    const float bd = best.x, bv = best.y, bw = best.z;
    const int   bf = __float_as_int(best.w);

    float4 q0 = ws[bf * 3 + 0];
    float4 q1 = ws[bf * 3 + 1];
    float4 q2 = ws[bf * 3 + 2];
    float cxp = q0.x + bv * q1.x + bw * q2.x;
    float cyp = q0.y + bv * q1.y + bw * q2.y;
    float czp = q0.z + bv * q1.z + bw * q2.z;

    // Outputs concatenated: dist2 [NPTS] f32 | closest [NPTS*3] f32 | faces [NPTS] i32
    out[pi] = bd;
    float* cp = out + NPTS;
    cp[3 * pi + 0] = cxp;
    cp[3 * pi + 1] = cyp;
    cp[3 * pi + 2] = czp;
    ((int*)(out + NPTS + NPTS * 3))[pi] = bf;
}

extern "C" void kernel_launch(void* const* d_in, const int* in_sizes, int n_in,
                              void* d_out, int out_size, void* d_ws, size_t ws_size,
                              hipStream_t stream) {
    const float* tri = (const float*)d_in[0];   // [1,1024,3,3] f32
    const float* pts = (const float*)d_in[1];   // [1,16384,3]  f32
    float* out = (float*)d_out;                 // 81920 elements
    float4* ws = (float4*)d_ws;                 // 48 KB table + 1 MB candidates

    tri_prep_kernel<<<(NTRI + 255) / 256, 256, 0, stream>>>(tri, ws);
    dim3 grid(NPTS / BLK, NSLICE);
    slice_kernel<<<grid, BLK, 0, stream>>>(pts, ws, ws + WS_CAND_BASE);
    reduce_kernel<<<(NPTS + 255) / 256, 256, 0, stream>>>(ws, out);
}